// Model2_26285199851849
// MI455X (gfx1250) — compile-verified
//
#include <hip/hip_runtime.h>

// ---- problem constants (from reference) ----
#define USER_N 100000
#define ITEM_N 150000
#define LATDIM 128
#define NNODES (USER_N + ITEM_N)   // 250000
#define NEDGES 2000000

typedef __attribute__((ext_vector_type(2))) float v2f;
typedef __attribute__((ext_vector_type(8))) float v8f;

// Full-precision fp32 WMMA: D(16x16) = A(16x4) * B(4x16) + C
__device__ __forceinline__ v8f wmma4(v2f a, v2f b, v8f c) {
  return __builtin_amdgcn_wmma_f32_16x16x4_f32(
      /*neg_a=*/false, a, /*neg_b=*/false, b,
      /*c_mod=*/(short)0, c, /*reuse_a=*/false, /*reuse_b=*/false);
}

__device__ __forceinline__ void atomAddF(float* p, float v) {
  __hip_atomic_fetch_add(p, v, __ATOMIC_RELAXED, __HIP_MEMORY_SCOPE_AGENT);
}

// ---------------------------------------------------------------------------
// zero-fill (atomic-accumulation targets must start at 0 every call)
// ---------------------------------------------------------------------------
__global__ void zero_f4(float4* __restrict__ p, long n4) {
  long i = (long)blockIdx.x * blockDim.x + threadIdx.x;
  long stride = (long)gridDim.x * blockDim.x;
  float4 z = make_float4(0.f, 0.f, 0.f, 0.f);
  for (; i < n4; i += stride) p[i] = z;
}

// ---------------------------------------------------------------------------
// G = X^T * X   (X: [rows,128] fp32, G: [128,128])
// Chunks of 64 rows staged in LDS; each of 8 waves owns one 16-row tile band
// of G (8 tiles of 16x16), accumulated in v8f regs, atomically reduced at end.
// ---------------------------------------------------------------------------
__global__ void __launch_bounds__(256) gram_kernel(const float* __restrict__ X,
                                                   int rows,
                                                   float* __restrict__ G) {
  __shared__ float xs[64][132];   // 64 rows x 128 cols, padded vs bank conflicts
  const int tid  = threadIdx.x;
  const int lane = tid & 31;
  const int w    = tid >> 5;      // wave 0..7 -> tile row band
  const int col  = lane & 15;
  const int hi   = lane >> 4;     // 0: lanes 0-15, 1: lanes 16-31
  const int kk   = hi * 2;        // K sub-offset per ISA fp32 A/B layout

  v8f acc[8];
#pragma unroll
  for (int b = 0; b < 8; ++b)
#pragma unroll
    for (int j = 0; j < 8; ++j) acc[b][j] = 0.0f;

  const int nchunks = (rows + 63) >> 6;
  for (int ch = blockIdx.x; ch < nchunks; ch += gridDim.x) {
    const int r0 = ch << 6;
    // cooperative stage: 64x128 floats (tail rows padded with zeros)
    for (int i = tid; i < 64 * 32; i += 256) {
      int rr = i >> 5;
      int c4 = (i & 31) << 2;
      float4 val;
      if (r0 + rr < rows)
        val = *(const float4*)(X + (size_t)(r0 + rr) * LATDIM + c4);
      else
        val = make_float4(0.f, 0.f, 0.f, 0.f);
      *(float4*)(&xs[rr][c4]) = val;
    }
    __syncthreads();

    for (int k = 0; k < 64; k += 4) {
      v2f a;   // A[i,k'] = X[r0+k+k', 16w+i]  (column band w, transposed)
      a.x = xs[k + kk][16 * w + col];
      a.y = xs[k + kk + 1][16 * w + col];
#pragma unroll
      for (int b = 0; b < 8; ++b) {
        v2f bb;  // B[k',j] = X[r0+k+k', 16b+j]
        bb.x = xs[k + kk][16 * b + col];
        bb.y = xs[k + kk + 1][16 * b + col];
        acc[b] = wmma4(a, bb, acc[b]);
      }
    }
    __syncthreads();
  }

  // reduce partial Gram into global G
#pragma unroll
  for (int b = 0; b < 8; ++b)
#pragma unroll
    for (int vv = 0; vv < 8; ++vv) {
      int m = 16 * w + vv + hi * 8;  // C/D layout: M = vgpr + 8*half
      int n = 16 * b + col;
      atomAddF(&G[m * 128 + n], acc[b][vv]);
    }
}

// ---------------------------------------------------------------------------
// T = H^T * G   then   W = H * T    (all 128x128, tiny)
// blockIdx.x selects user(0)/item(1) side.
// ---------------------------------------------------------------------------
__global__ void __launch_bounds__(256) hyperT_kernel(
    const float* __restrict__ uH, const float* __restrict__ iH,
    const float* __restrict__ Gu, const float* __restrict__ Gi,
    float* __restrict__ Tu, float* __restrict__ Ti) {
  const float* H = blockIdx.x ? iH : uH;
  const float* G = blockIdx.x ? Gi : Gu;
  float* T = blockIdx.x ? Ti : Tu;
  for (int o = threadIdx.x; o < 128 * 128; o += blockDim.x) {
    int h = o >> 7, d2 = o & 127;
    float s = 0.f;
    for (int d = 0; d < 128; ++d) s = fmaf(H[d * 128 + h], G[d * 128 + d2], s);
    T[o] = s;
  }
}

__global__ void __launch_bounds__(256) hyperW_kernel(
    const float* __restrict__ uH, const float* __restrict__ iH,
    const float* __restrict__ Tu, const float* __restrict__ Ti,
    float* __restrict__ Wu, float* __restrict__ Wi) {
  const float* H = blockIdx.x ? iH : uH;
  const float* T = blockIdx.x ? Ti : Tu;
  float* W = blockIdx.x ? Wi : Wu;
  for (int o = threadIdx.x; o < 128 * 128; o += blockDim.x) {
    int d = o >> 7, d2 = o & 127;
    float s = 0.f;
    for (int h = 0; h < 128; ++h) s = fmaf(H[d * 128 + h], T[h * 128 + d2], s);
    W[o] = s;
  }
}

// ---------------------------------------------------------------------------
// hyper = relu(X @ W) + X   written to BOTH layer slots of hyper_lats.
// One wave per 16-row block (USER boundary is 16-aligned: 100000 = 6250*16).
// ---------------------------------------------------------------------------
__global__ void __launch_bounds__(256) hyper_out_kernel(
    const float* __restrict__ u_emb, const float* __restrict__ i_emb,
    const float* __restrict__ Wu, const float* __restrict__ Wi,
    float* __restrict__ hyp0, float* __restrict__ hyp1) {
  const int lane = threadIdx.x & 31;
  const int col  = lane & 15;
  const int hi   = lane >> 4;
  const int kk   = hi * 2;
  const int nwaves = (gridDim.x * blockDim.x) >> 5;
  int wave = ((blockIdx.x * blockDim.x) + threadIdx.x) >> 5;
  const int nblk = NNODES / 16;  // 15625

  for (int blk = wave; blk < nblk; blk += nwaves) {
    const int r0 = blk * 16;
    const float* x;
    const float* W;
    if (r0 < USER_N) { x = u_emb + (size_t)r0 * 128;            W = Wu; }
    else             { x = i_emb + (size_t)(r0 - USER_N) * 128; W = Wi; }

    v8f acc[8];
#pragma unroll
    for (int b = 0; b < 8; ++b)
#pragma unroll
      for (int j = 0; j < 8; ++j) acc[b][j] = 0.0f;

    for (int k = 0; k < 128; k += 4) {
      v2f a;  // A[i,k'] = x[i, k+k'], i = lane&15
      a.x = x[col * 128 + k + kk];
      a.y = x[col * 128 + k + kk + 1];
#pragma unroll
      for (int b = 0; b < 8; ++b) {
        v2f bb;  // B[k',j] = W[k+k', 16b+j]
        bb.x = W[(k + kk) * 128 + 16 * b + col];
        bb.y = W[(k + kk + 1) * 128 + 16 * b + col];
        acc[b] = wmma4(a, bb, acc[b]);
      }
    }

#pragma unroll
    for (int b = 0; b < 8; ++b)
#pragma unroll
      for (int vv = 0; vv < 8; ++vv) {
        int m = vv + hi * 8;
        int n = 16 * b + col;
        float val = acc[b][vv];
        val = val > 0.f ? val : 0.f;      // relu
        val += x[m * 128 + n];            // residual
        size_t off = (size_t)(r0 + m) * 128 + n;
        hyp0[off] = val;
        hyp1[off] = val;
      }
  }
}

// ---------------------------------------------------------------------------
// SpMM: out[row] += mask(rand)*vals*1.25 * x[col], one wave per edge
// (32 lanes x float4 = 128 dims). x given as (u-part, i-part) pointers so the
// same kernel serves embeds0 (two tables) and lat1 (contiguous, split view).
// ---------------------------------------------------------------------------
__global__ void __launch_bounds__(256) spmm_kernel(
    const int* __restrict__ rows, const int* __restrict__ cols,
    const float* __restrict__ vals, const float* __restrict__ rnd,
    const float* __restrict__ xu, const float* __restrict__ xi,
    float* __restrict__ out) {
  const int lane = threadIdx.x & 31;
  const int nwaves = (gridDim.x * blockDim.x) >> 5;
  int wv = ((blockIdx.x * blockDim.x) + threadIdx.x) >> 5;

  for (int e = wv; e < NEDGES; e += nwaves) {
    float mask = floorf(rnd[e] + 0.8f);   // bit-exact with reference dropedge
    if (mask == 0.f) continue;
    int row = rows[e];
    int cc  = cols[e];
    float v = vals[e] * mask * 1.25f;     // 1/KEEP_RATE
    const float* xp = (cc < USER_N) ? (xu + (size_t)cc * 128)
                                    : (xi + (size_t)(cc - USER_N) * 128);
    float4 xv = *(const float4*)(xp + lane * 4);
    float* op = out + (size_t)row * 128 + lane * 4;
    atomAddF(op + 0, v * xv.x);
    atomAddF(op + 1, v * xv.y);
    atomAddF(op + 2, v * xv.z);
    atomAddF(op + 3, v * xv.w);
  }
}

// ---------------------------------------------------------------------------
// total = embeds0 + lat1 + lat2  (elementwise, float4)
// ---------------------------------------------------------------------------
__global__ void total_kernel(const float* __restrict__ u_emb,
                             const float* __restrict__ i_emb,
                             const float* __restrict__ lat1,
                             const float* __restrict__ lat2,
                             float* __restrict__ total) {
  const long n4 = (long)NNODES * (LATDIM / 4);
  const long usplit = (long)USER_N * (LATDIM / 4);
  long i = (long)blockIdx.x * blockDim.x + threadIdx.x;
  long stride = (long)gridDim.x * blockDim.x;
  for (; i < n4; i += stride) {
    float4 x0 = (i < usplit) ? ((const float4*)u_emb)[i]
                             : ((const float4*)i_emb)[i - usplit];
    float4 a = ((const float4*)lat1)[i];
    float4 b = ((const float4*)lat2)[i];
    float4 t;
    t.x = x0.x + a.x + b.x;
    t.y = x0.y + a.y + b.y;
    t.z = x0.z + a.z + b.z;
    t.w = x0.w + a.w + b.w;
    ((float4*)total)[i] = t;
  }
}

// ---------------------------------------------------------------------------
extern "C" void kernel_launch(void* const* d_in, const int* in_sizes, int n_in,
                              void* d_out, int out_size, void* d_ws, size_t ws_size,
                              hipStream_t stream) {
  const int*   adj_rows  = (const int*)d_in[0];
  const int*   adj_cols  = (const int*)d_in[1];
  const float* adj_vals  = (const float*)d_in[2];
  const float* edge_rand = (const float*)d_in[3];
  const float* u_emb     = (const float*)d_in[4];
  const float* i_emb     = (const float*)d_in[5];
  const float* u_hyper   = (const float*)d_in[6];
  const float* i_hyper   = (const float*)d_in[7];

  float* out = (float*)d_out;
  const size_t nd = (size_t)NNODES * LATDIM;
  float* total = out;            // [N,128]
  float* lat1  = out + nd;       // gnn_lats[0]
  float* lat2  = out + 2 * nd;   // gnn_lats[1]
  float* hyp0  = out + 3 * nd;   // hyper_lats[0]
  float* hyp1  = out + 4 * nd;   // hyper_lats[1]

  float* ws = (float*)d_ws;
  float* Gu = ws + 0 * 16384;
  float* Gi = ws + 1 * 16384;
  float* Tu = ws + 2 * 16384;
  float* Ti = ws + 3 * 16384;
  float* Wu = ws + 4 * 16384;
  float* Wi = ws + 5 * 16384;

  // zero atomic-accumulation targets (every call: graph replays must match)
  zero_f4<<<2048, 256, 0, stream>>>((float4*)lat1, (long)(2 * nd / 4));
  zero_f4<<<16, 256, 0, stream>>>((float4*)Gu, (long)(2 * 16384 / 4));

  // hyper branch: Gram -> T -> W -> relu(X@W)+X (written to both layer slots)
  gram_kernel<<<480, 256, 0, stream>>>(u_emb, USER_N, Gu);
  gram_kernel<<<480, 256, 0, stream>>>(i_emb, ITEM_N, Gi);
  hyperT_kernel<<<2, 256, 0, stream>>>(u_hyper, i_hyper, Gu, Gi, Tu, Ti);
  hyperW_kernel<<<2, 256, 0, stream>>>(u_hyper, i_hyper, Tu, Ti, Wu, Wi);
  hyper_out_kernel<<<1024, 256, 0, stream>>>(u_emb, i_emb, Wu, Wi, hyp0, hyp1);

  // GNN branch: two SpMM layers then total
  spmm_kernel<<<2048, 256, 0, stream>>>(adj_rows, adj_cols, adj_vals,
                                        edge_rand, u_emb, i_emb, lat1);
  spmm_kernel<<<2048, 256, 0, stream>>>(adj_rows, adj_cols, adj_vals,
                                        edge_rand + NEDGES,
                                        lat1, lat1 + (size_t)USER_N * 128, lat2);
  total_kernel<<<2048, 256, 0, stream>>>(u_emb, i_emb, lat1, lat2, total);
}